// PatchTransformer_myaff_44796508897536
// MI455X (gfx1250) — compile-verified
//
#include <hip/hip_runtime.h>
#include <hip/hip_bf16.h>

// ---------------------------------------------------------------------------
// PatchTransformer for MI455X (gfx1250, wave32).
//
// Pipeline (all on `stream`, scratch in d_ws only, deterministic per call):
//   K0 zero_ws        : zero scalar block + resize-weight matrices + padded img
//   K1 build_weights  : bilinear resize matrices Ry (416x304), RxT (304x416)
//   K2 pack_img       : adv_patch (3x300x300) -> zero-padded 3x304x304
//   K3 scalars        : threefry-style PRNG scalars + 8x8 homography solve
//   K4 wmma_gemm      : tmp = Ry @ img        (V_WMMA_F32_16X16X4_F32)
//   K5 wmma_gemm      : p0  = tmp @ RxT       (V_WMMA_F32_16X16X4_F32)
//   K6 reduce_graymean: LDS tree reduce + global_atomic_add_f32
//   K7 finalize_patch : brightness/contrast/saturation mixes -> p_final (2MB)
//   K8 patch_apply    : fused rotation-bilinear o perspective-bilinear + noise
//                       (p_final is L2-resident; 232MB store stream dominates)
// ---------------------------------------------------------------------------

#define S     416
#define SRC   300
#define KP    304            // K padded to multiple of 4 for 16x16x4 WMMA
#define NIMG  112            // B*L = 8*14

// workspace layout (float offsets); total ~1.95M floats (~7.8 MB)
#define OFF_SCAL 0u
#define OFF_RY   64u                                   // 416*304
#define OFF_RXT  (OFF_RY  + (unsigned)(S   * KP))      // 304*416
#define OFF_IMG  (OFF_RXT + (unsigned)(KP  * S ))      // 3*304*304
#define OFF_TMP  (OFF_IMG + (unsigned)(3 * KP * KP))   // 3*416*304
#define OFF_P0   (OFF_TMP + (unsigned)(3 * S  * KP))   // 3*416*416
#define OFF_PF   (OFF_P0  + (unsigned)(3 * S  * S ))   // 3*416*416

// scalar block indices
// [0]=b [1]=contrast [2]=sat [3]=cos [4]=sin [5]=1/sc [6]=ctr [7]=gray-sum
// [8..15]=homography h[0..7]

typedef __attribute__((ext_vector_type(2))) float v2f;
typedef __attribute__((ext_vector_type(8))) float v8f;

// ---------------------------------------------------------------------------
// K0: zero the regions that are accumulated into / partially written
__global__ void zero_ws(float* ws, unsigned n) {
  unsigned t = blockIdx.x * blockDim.x + threadIdx.x;
  if (t < n) ws[t] = 0.0f;
}

// K1: separable bilinear resize weights (jax.image.resize half-pixel conv.)
__global__ void build_weights(float* ws) {
  int i = blockIdx.x * blockDim.x + threadIdx.x;
  if (i >= S) return;
  float* Ry  = ws + OFF_RY;
  float* RxT = ws + OFF_RXT;
  const float scale = (float)SRC / (float)S;
  float sp = ((float)i + 0.5f) * scale - 0.5f;
  float f  = floorf(sp);
  int   j0 = (int)f;
  float w1 = sp - f, w0 = 1.0f - w1;
  int j0c = min(max(j0, 0), SRC - 1);
  int j1c = min(max(j0 + 1, 0), SRC - 1);
  // row-resize matrix Ry[i][k]  (this thread owns row i)
  Ry[(size_t)i * KP + j0c] += w0;
  Ry[(size_t)i * KP + j1c] += w1;
  // col-resize matrix transposed RxT[k][i]  (this thread owns column i)
  RxT[(size_t)j0c * S + i] += w0;
  RxT[(size_t)j1c * S + i] += w1;
}

// K2: copy 3x300x300 patch into zero-padded 3x304x304
__global__ void pack_img(const float* __restrict__ adv, float* ws) {
  int t = blockIdx.x * blockDim.x + threadIdx.x;
  if (t >= 3 * SRC * SRC) return;
  int x = t % SRC, y = (t / SRC) % SRC, c = t / (SRC * SRC);
  ws[OFF_IMG + (size_t)c * KP * KP + (size_t)y * KP + x] = adv[t];
}

// ---------------------------------------------------------------------------
// threefry2x32 (jax-style counter PRNG), used only in the 1-thread kernel
__device__ __forceinline__ void tf2x32(unsigned k0, unsigned k1,
                                       unsigned c0, unsigned c1,
                                       unsigned* o0, unsigned* o1) {
  const unsigned ks2 = k0 ^ k1 ^ 0x1BD11BDAu;
  unsigned ks[3] = {k0, k1, ks2};
  unsigned x0 = c0 + k0, x1 = c1 + k1;
  const int rotA[4] = {13, 15, 26, 6};
  const int rotB[4] = {17, 29, 16, 24};
  for (int blk = 0; blk < 5; ++blk) {
    const int* rot = (blk & 1) ? rotB : rotA;
    for (int r = 0; r < 4; ++r) {
      x0 += x1;
      x1 = (x1 << rot[r]) | (x1 >> (32 - rot[r]));
      x1 ^= x0;
    }
    x0 += ks[(blk + 1) % 3];
    x1 += ks[(blk + 2) % 3] + (unsigned)(blk + 1);
  }
  *o0 = x0; *o1 = x1;
}

__device__ __forceinline__ float tf_uniform(unsigned k0, unsigned k1,
                                            float lo, float hi) {
  unsigned b0, b1;
  tf2x32(k0, k1, 0u, 0u, &b0, &b1);
  float f = __uint_as_float((b0 >> 9) | 0x3f800000u) - 1.0f;
  return lo + f * (hi - lo);
}

// K3: all scalar randomness + homography solve (one thread)
__global__ void scalars_kernel(float* ws) {
  if (blockIdx.x != 0 || threadIdx.x != 0) return;
  float* sc = ws + OFF_SCAL;

  // jax.random.key(42) -> (0,42); split into 7 subkeys
  unsigned key0 = 0u, key1 = 42u;
  unsigned sk0[7], sk1[7];
  for (int i = 0; i < 7; ++i)
    tf2x32(key0, key1, (unsigned)(2 * i), (unsigned)(2 * i + 1), &sk0[i], &sk1[i]);
  // kb kc ks kn ka ksc kp = subkeys 0..6
  float b  = tf_uniform(sk0[0], sk1[0], 0.8f, 1.1f);
  float cc = tf_uniform(sk0[1], sk1[1], 0.8f, 1.1f);
  float ss = tf_uniform(sk0[2], sk1[2], 0.8f, 1.1f);
  float th = tf_uniform(sk0[4], sk1[4], -45.0f, 45.0f) * (3.14159265358979f / 180.0f);
  float scl = tf_uniform(sk0[5], sk1[5], 0.1f, 0.16f);

  // 8 random corner perturbations in [0, dmax], dmax = int(0.6*(S//2)) = 124
  const unsigned dmaxp1 = 125u;
  float r[8];
  for (int j = 0; j < 8; ++j) {
    unsigned b0, b1;
    tf2x32(sk0[6], sk1[6], (unsigned)j, 0u, &b0, &b1);
    r[j] = (float)(b0 % dmaxp1);
  }
  const float w = (float)(S - 1);
  float ex[4] = {r[0], w - r[2], w - r[4], r[6]};
  float ey[4] = {r[1], r[3], w - r[5], w - r[7]};
  float px[4] = {0.0f, w, w, 0.0f};
  float py[4] = {0.0f, 0.0f, w, w};

  // assemble 8x8 system, Gaussian elimination w/ partial pivoting
  float A[8][9];
  for (int i = 0; i < 8; ++i)
    for (int j = 0; j < 9; ++j) A[i][j] = 0.0f;
  for (int i = 0; i < 4; ++i) {
    A[i][0] = ex[i]; A[i][1] = ey[i]; A[i][2] = 1.0f;
    A[i][6] = -px[i] * ex[i]; A[i][7] = -px[i] * ey[i]; A[i][8] = px[i];
    A[i + 4][3] = ex[i]; A[i + 4][4] = ey[i]; A[i + 4][5] = 1.0f;
    A[i + 4][6] = -py[i] * ex[i]; A[i + 4][7] = -py[i] * ey[i]; A[i + 4][8] = py[i];
  }
  for (int col = 0; col < 8; ++col) {
    int piv = col;
    for (int rr = col + 1; rr < 8; ++rr)
      if (fabsf(A[rr][col]) > fabsf(A[piv][col])) piv = rr;
    if (piv != col)
      for (int j = 0; j < 9; ++j) { float t = A[col][j]; A[col][j] = A[piv][j]; A[piv][j] = t; }
    float inv = 1.0f / A[col][col];
    for (int j = col; j < 9; ++j) A[col][j] *= inv;
    for (int rr = 0; rr < 8; ++rr) {
      if (rr == col) continue;
      float f = A[rr][col];
      if (f != 0.0f)
        for (int j = col; j < 9; ++j) A[rr][j] -= f * A[col][j];
    }
  }

  sc[0] = b;  sc[1] = cc; sc[2] = ss;
  sc[3] = cosf(th); sc[4] = sinf(th);
  sc[5] = 1.0f / scl; sc[6] = (float)(S - 1) * 0.5f;
  sc[7] = 0.0f;                       // gray-sum accumulator
  for (int i = 0; i < 8; ++i) sc[8 + i] = A[i][8];
}

// ---------------------------------------------------------------------------
// K4/K5: f32 GEMM on the matrix core. One wave32 per 16x16 C tile, K step 4.
// A-frag (16x4), B-frag (4x16), C/D per ISA 7.12.2 f32 layouts.
__global__ void wmma_gemm_f32(const float* __restrict__ A,
                              const float* __restrict__ B,
                              float* __restrict__ C,
                              int K, int lda, int ldb, int ldc,
                              long sA, long sB, long sC) {
  const int lane = threadIdx.x & 31;
  const int half = lane >> 4;     // 0: K pair {k,k+1}, 1: {k+2,k+3}
  const int l15  = lane & 15;
  const int ch   = blockIdx.z;
  A += (long)ch * sA; B += (long)ch * sB; C += (long)ch * sC;

  const int arow = blockIdx.x * 16 + l15;   // A row held by this lane
  const int bcol = blockIdx.y * 16 + l15;   // B col held by this lane

  v8f acc = {};
  for (int k = 0; k < K; k += 4) {
    v2f a, bb;
    a[0]  = A[(size_t)arow * lda + (k + 2 * half + 0)];
    a[1]  = A[(size_t)arow * lda + (k + 2 * half + 1)];
    bb[0] = B[(size_t)(k + 2 * half + 0) * ldb + bcol];
    bb[1] = B[(size_t)(k + 2 * half + 1) * ldb + bcol];
    acc = __builtin_amdgcn_wmma_f32_16x16x4_f32(
        false, a, false, bb, (short)0, acc, false, false);
  }
  const int crow0 = blockIdx.x * 16 + half * 8;   // lanes 16-31 hold M=r+8
  const int ccol  = blockIdx.y * 16 + l15;
#pragma unroll
  for (int r = 0; r < 8; ++r)
    C[(size_t)(crow0 + r) * ldc + ccol] = acc[r];
}

// ---------------------------------------------------------------------------
// K6: sum over pixels of dot(GRAY_W, b * p0[:,pix])
__global__ void reduce_graymean(const float* __restrict__ p0, float* scal) {
  __shared__ float sdata[256];
  int pix = blockIdx.x * blockDim.x + threadIdx.x;
  float v = 0.0f;
  if (pix < S * S) {
    float b = scal[0];
    v = b * (0.2989f * p0[pix] + 0.587f * p0[S * S + pix] + 0.114f * p0[2 * S * S + pix]);
  }
  sdata[threadIdx.x] = v;
  __syncthreads();
  for (int s2 = 128; s2 > 0; s2 >>= 1) {
    if ((int)threadIdx.x < s2) sdata[threadIdx.x] += sdata[threadIdx.x + s2];
    __syncthreads();
  }
  if (threadIdx.x == 0) atomicAdd(&scal[7], sdata[0]);  // global_atomic_add_f32
}

// K7: contrast + saturation mixes -> p_final (2 MB, stays in L2)
__global__ void finalize_patch(const float* __restrict__ p0,
                               float* __restrict__ pf,
                               const float* __restrict__ scal) {
  int pix = blockIdx.x * blockDim.x + threadIdx.x;
  if (pix >= S * S) return;
  float b = scal[0], cc = scal[1], ss = scal[2];
  float m = scal[7] * (1.0f / (float)(S * S));
  float p0v = cc * (b * p0[pix])             + (1.0f - cc) * m;
  float p1v = cc * (b * p0[S * S + pix])     + (1.0f - cc) * m;
  float p2v = cc * (b * p0[2 * S * S + pix]) + (1.0f - cc) * m;
  float g = 0.2989f * p0v + 0.587f * p1v + 0.114f * p2v;
  pf[pix]             = ss * p0v + (1.0f - ss) * g;
  pf[S * S + pix]     = ss * p1v + (1.0f - ss) * g;
  pf[2 * S * S + pix] = ss * p2v + (1.0f - ss) * g;
}

// ---------------------------------------------------------------------------
// K8: fused rotation-bilinear o perspective-bilinear with per-image noise
__device__ __forceinline__ float noise_u(unsigned idx) {
  unsigned h = idx * 0x9E3779B9u;
  h ^= h >> 16; h *= 0x85EBCA6Bu;
  h ^= h >> 13; h *= 0xC2B2AE35u;
  h ^= h >> 16;
  return (float)(h >> 8) * (2.0f / 16777216.0f) - 1.0f;
}

__device__ __forceinline__ float batch_val(const float* __restrict__ pf,
                                           int n, int c, int yy, int xx) {
  if (xx < 0 || xx > S - 1 || yy < 0 || yy > S - 1) return 0.0f;  // zero pad
  float v = pf[((size_t)c * S + yy) * S + xx];
  unsigned idx = (((unsigned)n * 3u + (unsigned)c) * (unsigned)S + (unsigned)yy)
                 * (unsigned)S + (unsigned)xx;
  return v + 0.15f * noise_u(idx);
}

__device__ __forceinline__ float affine_sample(const float* __restrict__ pf,
                                               int n, int c, int Y, int X,
                                               float cos_t, float sin_t,
                                               float inv_sc, float ctr) {
  if (X < 0 || X > S - 1 || Y < 0 || Y > S - 1) return 0.0f;  // zero pad
  float dx = (float)X - ctr, dy = (float)Y - ctr;
  float sx = (cos_t * dx + sin_t * dy) * inv_sc + ctr;
  float sy = (-sin_t * dx + cos_t * dy) * inv_sc + ctr;
  float fx = floorf(sx), fy = floorf(sy);
  int x0 = (int)fx, y0 = (int)fy;
  float wx1 = sx - fx, wy1 = sy - fy;
  float wx0 = 1.0f - wx1, wy0 = 1.0f - wy1;
  return batch_val(pf, n, c, y0, x0)         * (wy0 * wx0)
       + batch_val(pf, n, c, y0, x0 + 1)     * (wy0 * wx1)
       + batch_val(pf, n, c, y0 + 1, x0)     * (wy1 * wx0)
       + batch_val(pf, n, c, y0 + 1, x0 + 1) * (wy1 * wx1);
}

__global__ void patch_apply(const float* __restrict__ pf,
                            const float* __restrict__ scal,
                            float* __restrict__ out) {
  __builtin_prefetch(pf, 0, 0);  // global_prefetch_b8: pull patch toward L2
  unsigned t = blockIdx.x * blockDim.x + threadIdx.x;
  const unsigned total = (unsigned)NIMG * 3u * S * S;
  if (t >= total) return;
  int x = (int)(t % S);
  unsigned t1 = t / S;
  int y = (int)(t1 % S);
  unsigned t2 = t1 / S;
  int c = (int)(t2 % 3u);
  int n = (int)(t2 / 3u);

  float cos_t = scal[3], sin_t = scal[4], inv_sc = scal[5], ctr = scal[6];
  float h0 = scal[8],  h1 = scal[9],  h2 = scal[10], h3 = scal[11];
  float h4 = scal[12], h5 = scal[13], h6 = scal[14], h7 = scal[15];

  float xo = (float)x + 0.5f, yo = (float)y + 0.5f;
  float inv_den = 1.0f / (h6 * xo + h7 * yo + 1.0f);
  float psx = (h0 * xo + h1 * yo + h2) * inv_den - 0.5f;
  float psy = (h3 * xo + h4 * yo + h5) * inv_den - 0.5f;

  float fx = floorf(psx), fy = floorf(psy);
  int X0 = (int)fx, Y0 = (int)fy;
  float wx1 = psx - fx, wy1 = psy - fy;
  float wx0 = 1.0f - wx1, wy0 = 1.0f - wy1;

  float v = affine_sample(pf, n, c, Y0,     X0,     cos_t, sin_t, inv_sc, ctr) * (wy0 * wx0)
          + affine_sample(pf, n, c, Y0,     X0 + 1, cos_t, sin_t, inv_sc, ctr) * (wy0 * wx1)
          + affine_sample(pf, n, c, Y0 + 1, X0,     cos_t, sin_t, inv_sc, ctr) * (wy1 * wx0)
          + affine_sample(pf, n, c, Y0 + 1, X0 + 1, cos_t, sin_t, inv_sc, ctr) * (wy1 * wx1);

  out[t] = fminf(fmaxf(v, 0.0f), 0.99f);
}

// ---------------------------------------------------------------------------
extern "C" void kernel_launch(void* const* d_in, const int* in_sizes, int n_in,
                              void* d_out, int out_size, void* d_ws, size_t ws_size,
                              hipStream_t stream) {
  (void)in_sizes; (void)n_in; (void)out_size; (void)ws_size;
  const float* adv = (const float*)d_in[0];   // 3x300x300 f32
  // d_in[1] (lab_batch) only contributes shapes; d_in[2] (img_size=416) hardcoded.
  float* out = (float*)d_out;                 // 8x14x3x416x416 f32
  float* ws  = (float*)d_ws;                  // needs ~7.8 MB

  // K0: zero accumulated regions (scalars + Ry + RxT + padded img)
  const unsigned zn = OFF_TMP;
  zero_ws<<<(zn + 255u) / 256u, 256, 0, stream>>>(ws, zn);
  // K1/K2
  build_weights<<<(S + 63) / 64, 64, 0, stream>>>(ws);
  pack_img<<<(3 * SRC * SRC + 255) / 256, 256, 0, stream>>>(adv, ws);
  // K3
  scalars_kernel<<<1, 1, 0, stream>>>(ws);
  // K4: tmp(416x304) = Ry(416x304) @ img(304x304), per channel
  wmma_gemm_f32<<<dim3(S / 16, KP / 16, 3), 32, 0, stream>>>(
      ws + OFF_RY, ws + OFF_IMG, ws + OFF_TMP,
      KP, KP, KP, KP, 0L, (long)(KP * KP), (long)(S * KP));
  // K5: p0(416x416) = tmp(416x304) @ RxT(304x416), per channel
  wmma_gemm_f32<<<dim3(S / 16, S / 16, 3), 32, 0, stream>>>(
      ws + OFF_TMP, ws + OFF_RXT, ws + OFF_P0,
      KP, KP, S, S, (long)(S * KP), 0L, (long)(S * S));
  // K6/K7
  reduce_graymean<<<(S * S + 255) / 256, 256, 0, stream>>>(ws + OFF_P0, ws + OFF_SCAL);
  finalize_patch<<<(S * S + 255) / 256, 256, 0, stream>>>(ws + OFF_P0, ws + OFF_PF,
                                                          ws + OFF_SCAL);
  // K8: 112*3*416*416 outputs
  const unsigned total = (unsigned)NIMG * 3u * S * S;
  patch_apply<<<(total + 255u) / 256u, 256, 0, stream>>>(ws + OFF_PF, ws + OFF_SCAL, out);
}